// DGCNN_171798692567
// MI455X (gfx1250) — compile-verified
//
#include <hip/hip_runtime.h>
#include <hip/hip_bf16.h>

#define NPTS  16384
#define BATCH 8
#define NPG   2048
#define KNBR  20
#define OUTC  16

typedef __attribute__((ext_vector_type(2))) float v2f;
typedef __attribute__((ext_vector_type(8))) float v8f;

static __device__ __forceinline__ v8f wmma_f32_k4(v2f a, v2f b, v8f c) {
  // D = A(16x4 f32) * B(4x16 f32) + C(16x16 f32)
  return __builtin_amdgcn_wmma_f32_16x16x4_f32(false, a, false, b, (short)0, c, false, false);
}

// CDNA5 async global->LDS copy (ASYNCcnt-tracked), 16 bytes per lane.
// LDS address = low 32 bits of the generic pointer (LDS aperture truncation, ISA 10.2).
static __device__ __forceinline__ void async_b128(void* lds_ptr, const void* gptr) {
  unsigned l = (unsigned)(size_t)lds_ptr;
  asm volatile("global_load_async_to_lds_b128 %0, %1, off" :: "v"(l), "v"(gptr) : "memory");
}
static __device__ __forceinline__ void wait_async_zero() {
  asm volatile("s_wait_asynccnt 0x0" ::: "memory");
}

// ---------------------------------------------------------------- init
__global__ __launch_bounds__(256) void init_kernel(float* stats1, float* stats2,
                                                   float* statsf, unsigned* fmaxp,
                                                   unsigned* fminp) {
  int i = blockIdx.x * 256 + threadIdx.x;
  if (i < 128) stats1[i] = 0.f;
  if (i < 256) stats2[i] = 0.f;
  if (i < 512) statsf[i] = 0.f;
  if (i < BATCH * 256) { fmaxp[i] = 0u; fminp[i] = 0x7F800000u; }  // relu>=0; min=+inf
}

// ---------------------------------------------------------------- f0 = [pos|x], sq
__global__ __launch_bounds__(256) void f0_kernel(const float* pos, const float* x,
                                                 float* f0, float* sq) {
  int n = blockIdx.x * 256 + threadIdx.x;
  if (n >= NPTS) return;
  float a = pos[n * 3 + 0], b = pos[n * 3 + 1], c = pos[n * 3 + 2], d = x[n];
  f0[n * 4 + 0] = a; f0[n * 4 + 1] = b; f0[n * 4 + 2] = c; f0[n * 4 + 3] = d;
  sq[n] = a * a + b * b + c * c + d * d;
}

// ---------------------------------------------------------------- KNN #1 (global, 4-D)
// one wave per query, 8 queries/block; double-buffered async candidate tiles
__global__ __launch_bounds__(256) void knn1_kernel(const float* f0, const float* sq,
                                                   int* idx1) {
  union SharedU1 {
    struct { float cf[2][4096]; float csq[2][1024]; } st;      // 2 x (16KB + 4KB)
    struct { float md[8][32][KNBR]; int mi[8][32][KNBR]; } mg; // 40KB merge scratch
  };
  __shared__ __align__(16) SharedU1 sh;
  int tid = threadIdx.x;
  int wave = tid >> 5, lane = tid & 31;
  int q = blockIdx.x * 8 + wave;
  float4 qf = ((const float4*)f0)[q];
  float dist[KNBR]; int idxl[KNBR];
#pragma unroll
  for (int t = 0; t < KNBR; ++t) { dist[t] = 1e30f; idxl[t] = 0x7FFFFFFF; }
  // prologue: stage chunk 0 into buffer 0
  for (int e = tid; e < 1024; e += 256)
    async_b128(&sh.st.cf[0][e * 4], &f0[(0 + e) * 4]);
  async_b128(&sh.st.csq[0][tid * 4], &sq[tid * 4]);
  for (int chunk = 0; chunk < NPTS / 1024; ++chunk) {
    int b = chunk & 1;
    int base = chunk * 1024;
    wait_async_zero();     // own async loads (chunk `chunk`) have landed in LDS
    __syncthreads();       // every wave's loads landed & previous buffer free
    if (chunk + 1 < NPTS / 1024) {
      int nbase = base + 1024;
      for (int e = tid; e < 1024; e += 256)
        async_b128(&sh.st.cf[b ^ 1][e * 4], &f0[(nbase + e) * 4]);
      async_b128(&sh.st.csq[b ^ 1][tid * 4], &sq[nbase + tid * 4]);
    }
    for (int i = 0; i < 32; ++i) {
      int cl = lane + 32 * i;
      float4 c = *(const float4*)&sh.st.cf[b][cl * 4];
      float dot = qf.x * c.x + qf.y * c.y + qf.z * c.z + qf.w * c.w;
      float d = sh.st.csq[b][cl] - 2.f * dot;  // + sq_q is row-constant: ordering unchanged
      if (d < dist[KNBR - 1]) {
        dist[KNBR - 1] = d; idxl[KNBR - 1] = base + cl;
#pragma unroll
        for (int t = KNBR - 1; t > 0; --t) {
          if (dist[t] < dist[t - 1]) {
            float td = dist[t]; dist[t] = dist[t - 1]; dist[t - 1] = td;
            int ti = idxl[t]; idxl[t] = idxl[t - 1]; idxl[t - 1] = ti;
          }
        }
      }
    }
  }
  __syncthreads();  // staging done; reuse LDS as merge scratch
#pragma unroll
  for (int t = 0; t < KNBR; ++t) { sh.mg.md[wave][lane][t] = dist[t]; sh.mg.mi[wave][lane][t] = idxl[t]; }
  __syncthreads();
  int p = 0;
  for (int r = 0; r < KNBR; ++r) {
    float bd = (p < KNBR) ? sh.mg.md[wave][lane][p] : 1e30f;
    int   bi = (p < KNBR) ? sh.mg.mi[wave][lane][p] : 0x7FFFFFFF;
    int   bl = lane;
#pragma unroll
    for (int off = 16; off >= 1; off >>= 1) {
      float od = __shfl_xor(bd, off, 32);
      int   oi = __shfl_xor(bi, off, 32);
      int   ol = __shfl_xor(bl, off, 32);
      if (od < bd || (od == bd && oi < bi)) { bd = od; bi = oi; bl = ol; }
    }
    if (lane == bl) p++;
    if (lane == 0) idx1[q * KNBR + r] = bi;
  }
}

// ---------------------------------------------------------------- MLP1 (8->64), stats + max/min over K
__global__ __launch_bounds__(256) void mlp1_kernel(const float* f0, const int* idx1,
                                                   const float* w1, const float* b1,
                                                   float* z1max, float* z1min, float* stats1) {
  __shared__ float lsum[64], lsq[64];
  if (threadIdx.x < 64) { lsum[threadIdx.x] = 0.f; lsq[threadIdx.x] = 0.f; }
  __syncthreads();
  int n = blockIdx.x * 8 + (threadIdx.x >> 5);
  int lane = threadIdx.x & 31;
  float xi0 = f0[n * 4 + 0], xi1 = f0[n * 4 + 1], xi2 = f0[n * 4 + 2], xi3 = f0[n * 4 + 3];
#pragma unroll
  for (int cc = 0; cc < 2; ++cc) {
    int c = lane + 32 * cc;
    float wc[8];
#pragma unroll
    for (int d = 0; d < 8; ++d) wc[d] = w1[d * 64 + c];
    float base = b1[c] + xi0 * wc[0] + xi1 * wc[1] + xi2 * wc[2] + xi3 * wc[3];
    float s = 0.f, ss = 0.f, mx = -1e30f, mn = 1e30f;
    for (int k = 0; k < KNBR; ++k) {
      int j = idx1[n * KNBR + k];
      float z = base + (f0[j * 4 + 0] - xi0) * wc[4] + (f0[j * 4 + 1] - xi1) * wc[5] +
                (f0[j * 4 + 2] - xi2) * wc[6] + (f0[j * 4 + 3] - xi3) * wc[7];
      z = z > 0.f ? z : 0.f;
      s += z; ss += z * z; mx = fmaxf(mx, z); mn = fminf(mn, z);
    }
    z1max[n * 64 + c] = mx; z1min[n * 64 + c] = mn;
    atomicAdd(&lsum[c], s); atomicAdd(&lsq[c], ss);
  }
  __syncthreads();
  if (threadIdx.x < 64) {
    atomicAdd(&stats1[threadIdx.x], lsum[threadIdx.x]);
    atomicAdd(&stats1[64 + threadIdx.x], lsq[threadIdx.x]);
  }
}

// ---------------------------------------------------------------- BN1 finalize -> x1, row norms
__global__ __launch_bounds__(256) void bn1_kernel(const float* z1max, const float* z1min,
                                                  const float* stats1, const float* g1,
                                                  const float* be1, float* x1, float* x1sq) {
  int n = blockIdx.x * 8 + (threadIdx.x >> 5);
  int lane = threadIdx.x & 31;
  const float cnt = (float)NPTS * (float)KNBR;
  float ssq = 0.f;
#pragma unroll
  for (int cc = 0; cc < 2; ++cc) {
    int c = lane + 32 * cc;
    float m = stats1[c] / cnt;
    float v = stats1[64 + c] / cnt - m * m;
    float sc = g1[c] * rsqrtf(v + 1e-5f);
    float sh = be1[c] - m * sc;
    float val = sc > 0.f ? sc * z1max[n * 64 + c] + sh : sc * z1min[n * 64 + c] + sh;
    x1[n * 64 + c] = val;
    ssq += val * val;
  }
#pragma unroll
  for (int off = 16; off >= 1; off >>= 1) ssq += __shfl_xor(ssq, off, 32);
  if (lane == 0) x1sq[n] = ssq;
}

// ---------------------------------------------------------------- KNN #2 (per batch, 64-D)
// double-buffered async candidate tiles, 64 candidates x 64 feats per chunk
__global__ __launch_bounds__(256) void knn2_kernel(const float* x1, const float* x1sq,
                                                   int* idx2) {
  union SharedU2 {
    struct { float cf[2][64][64]; float csq[2][64]; } st;      // 2 x (16KB + 256B)
    struct { float md[8][32][KNBR]; int mi[8][32][KNBR]; } mg; // 40KB merge scratch
  };
  __shared__ __align__(16) SharedU2 sh;
  __shared__ float qfeat[8][64];
  int tid = threadIdx.x;
  int wave = tid >> 5, lane = tid & 31;
  int q = blockIdx.x * 8 + wave;
  int cbase0 = (q >> 11) << 11;  // batch * 2048
  for (int d = lane; d < 64; d += 32) qfeat[wave][d] = x1[q * 64 + d];
  float dist[KNBR]; int idxl[KNBR];
#pragma unroll
  for (int t = 0; t < KNBR; ++t) { dist[t] = 1e30f; idxl[t] = 0x7FFFFFFF; }
  // prologue: stage chunk 0 into buffer 0
  for (int e = tid; e < 1024; e += 256)
    async_b128(&sh.st.cf[0][e >> 4][(e & 15) * 4], &x1[(cbase0 + (e >> 4)) * 64 + (e & 15) * 4]);
  if (tid < 16) async_b128(&sh.st.csq[0][tid * 4], &x1sq[cbase0 + tid * 4]);
  for (int chunk = 0; chunk < NPG / 64; ++chunk) {
    int b = chunk & 1;
    int base = cbase0 + chunk * 64;
    wait_async_zero();
    __syncthreads();
    if (chunk + 1 < NPG / 64) {
      int nbase = base + 64;
      for (int e = tid; e < 1024; e += 256)
        async_b128(&sh.st.cf[b ^ 1][e >> 4][(e & 15) * 4], &x1[(nbase + (e >> 4)) * 64 + (e & 15) * 4]);
      if (tid < 16) async_b128(&sh.st.csq[b ^ 1][tid * 4], &x1sq[nbase + tid * 4]);
    }
#pragma unroll
    for (int i = 0; i < 2; ++i) {
      int cl = lane + 32 * i;
      float dot = 0.f;
      for (int d = 0; d < 64; ++d) dot += qfeat[wave][d] * sh.st.cf[b][cl][d];
      float dd = sh.st.csq[b][cl] - 2.f * dot;
      if (dd < dist[KNBR - 1]) {
        dist[KNBR - 1] = dd; idxl[KNBR - 1] = base + cl;  // GLOBAL index
#pragma unroll
        for (int t = KNBR - 1; t > 0; --t) {
          if (dist[t] < dist[t - 1]) {
            float td = dist[t]; dist[t] = dist[t - 1]; dist[t - 1] = td;
            int ti = idxl[t]; idxl[t] = idxl[t - 1]; idxl[t - 1] = ti;
          }
        }
      }
    }
  }
  __syncthreads();  // staging done; reuse LDS as merge scratch
#pragma unroll
  for (int t = 0; t < KNBR; ++t) { sh.mg.md[wave][lane][t] = dist[t]; sh.mg.mi[wave][lane][t] = idxl[t]; }
  __syncthreads();
  int p = 0;
  for (int r = 0; r < KNBR; ++r) {
    float bd = (p < KNBR) ? sh.mg.md[wave][lane][p] : 1e30f;
    int   bi = (p < KNBR) ? sh.mg.mi[wave][lane][p] : 0x7FFFFFFF;
    int   bl = lane;
#pragma unroll
    for (int off = 16; off >= 1; off >>= 1) {
      float od = __shfl_xor(bd, off, 32);
      int   oi = __shfl_xor(bi, off, 32);
      int   ol = __shfl_xor(bl, off, 32);
      if (od < bd || (od == bd && oi < bi)) { bd = od; bi = oi; bl = ol; }
    }
    if (lane == bl) p++;
    if (lane == 0) idx2[q * KNBR + r] = bi;
  }
}

// ---------------------------------------------------------------- MLP2 (128->128) via WMMA f32 16x16x4
// one block per point; 20 edge rows (padded to 32) x 128 cols, K-contraction 128
__global__ __launch_bounds__(256) void mlp2_kernel(const float* x1, const int* idx2,
                                                   const float* w2, const float* b2,
                                                   float* z2max, float* z2min, float* stats2) {
  __shared__ float e2[32][128];
  __shared__ float lsum[128], lsq[128];
  int n = blockIdx.x;
  int tid = threadIdx.x;
  for (int e = tid; e < KNBR * 64; e += 256) {
    int k = e >> 6, c = e & 63;
    int j = idx2[n * KNBR + k];
    float xi = x1[n * 64 + c];
    e2[k][c] = xi;
    e2[k][64 + c] = x1[j * 64 + c] - xi;
  }
  for (int e = tid; e < (32 - KNBR) * 128; e += 256) e2[KNBR + (e >> 7)][e & 127] = 0.f;
  __syncthreads();
  int wave = tid >> 5, lane = tid & 31;
  int half = lane >> 4, arow = lane & 15;
  int col = wave * 16 + arow;  // 8 waves cover 128 output cols
  v8f accA = {}; v8f accB = {};
  for (int kk = 0; kk < 128; kk += 4) {
    int ks = kk + (half << 1);
    v2f a0, a1, b;
    a0[0] = e2[arow][ks];      a0[1] = e2[arow][ks + 1];
    a1[0] = e2[16 + arow][ks]; a1[1] = e2[16 + arow][ks + 1];
    b[0] = w2[ks * 128 + col]; b[1] = w2[(ks + 1) * 128 + col];
    accA = wmma_f32_k4(a0, b, accA);
    accB = wmma_f32_k4(a1, b, accB);
  }
  float bias = b2[col];
  float s = 0.f, ss = 0.f, mx = -1e30f, mn = 1e30f;
#pragma unroll
  for (int r = 0; r < 8; ++r) {  // rows r+8*half : all < 16, valid
    float z = accA[r] + bias; z = z > 0.f ? z : 0.f;
    s += z; ss += z * z; mx = fmaxf(mx, z); mn = fminf(mn, z);
  }
#pragma unroll
  for (int r = 0; r < 8; ++r) {  // rows 16 + r + 8*half : valid iff half==0 && r<4
    float z = accB[r] + bias; z = z > 0.f ? z : 0.f;
    if (half == 0 && r < (KNBR - 16)) { s += z; ss += z * z; mx = fmaxf(mx, z); mn = fminf(mn, z); }
  }
  s  += __shfl_xor(s, 16, 32);
  ss += __shfl_xor(ss, 16, 32);
  mx = fmaxf(mx, __shfl_xor(mx, 16, 32));
  mn = fminf(mn, __shfl_xor(mn, 16, 32));
  if (half == 0) {
    z2max[n * 128 + col] = mx; z2min[n * 128 + col] = mn;
    lsum[col] = s; lsq[col] = ss;
  }
  __syncthreads();
  if (tid < 128) { atomicAdd(&stats2[tid], lsum[tid]); atomicAdd(&stats2[128 + tid], lsq[tid]); }
}

// ---------------------------------------------------------------- BN2 finalize -> x2
__global__ __launch_bounds__(256) void bn2_kernel(const float* z2max, const float* z2min,
                                                  const float* stats2, const float* g2,
                                                  const float* be2, float* x2) {
  int i = blockIdx.x * 256 + threadIdx.x;
  int c = i & 127;
  const float cnt = (float)NPTS * (float)KNBR;
  float m = stats2[c] / cnt;
  float v = stats2[128 + c] / cnt - m * m;
  float sc = g2[c] * rsqrtf(v + 1e-5f);
  float sh = be2[c] - m * sc;
  x2[i] = sc > 0.f ? sc * z2max[i] + sh : sc * z2min[i] + sh;
}

// ---------------------------------------------------------------- MLPf (192->256) via WMMA + batch max/min
__global__ __launch_bounds__(256) void mlpf_kernel(const float* x1, const float* x2,
                                                   const float* wf, const float* bf,
                                                   float* statsf, unsigned* fmaxp,
                                                   unsigned* fminp) {
  __shared__ float ft[32][192];
  int r0 = blockIdx.x * 32;
  int tid = threadIdx.x;
  for (int e = tid; e < 32 * 192; e += 256) {
    int r = e / 192, d = e % 192;
    int n = r0 + r;
    ft[r][d] = (d < 64) ? x1[n * 64 + d] : x2[n * 128 + d - 64];
  }
  __syncthreads();
  int wave = tid >> 5, lane = tid & 31;
  int half = lane >> 4, arow = lane & 15;
  int batch = r0 >> 11;
  int col0 = (2 * wave) * 16 + arow;
  int col1 = col0 + 16;
  v8f a00 = {}, a01 = {}, a10 = {}, a11 = {};
  for (int kk = 0; kk < 192; kk += 4) {
    int ks = kk + (half << 1);
    v2f a0, a1, b0, b1;
    a0[0] = ft[arow][ks];      a0[1] = ft[arow][ks + 1];
    a1[0] = ft[16 + arow][ks]; a1[1] = ft[16 + arow][ks + 1];
    b0[0] = wf[ks * 256 + col0]; b0[1] = wf[(ks + 1) * 256 + col0];
    b1[0] = wf[ks * 256 + col1]; b1[1] = wf[(ks + 1) * 256 + col1];
    a00 = wmma_f32_k4(a0, b0, a00);
    a01 = wmma_f32_k4(a0, b1, a01);
    a10 = wmma_f32_k4(a1, b0, a10);
    a11 = wmma_f32_k4(a1, b1, a11);
  }
#pragma unroll
  for (int nt = 0; nt < 2; ++nt) {
    int col = nt ? col1 : col0;
    float bias = bf[col];
    float s = 0.f, ss = 0.f, mx = -1e30f, mn = 1e30f;
#pragma unroll
    for (int r = 0; r < 8; ++r) {
      float z = (nt ? a01[r] : a00[r]) + bias; z = z > 0.f ? z : 0.f;
      s += z; ss += z * z; mx = fmaxf(mx, z); mn = fminf(mn, z);
      z = (nt ? a11[r] : a10[r]) + bias; z = z > 0.f ? z : 0.f;
      s += z; ss += z * z; mx = fmaxf(mx, z); mn = fminf(mn, z);
    }
    s  += __shfl_xor(s, 16, 32);
    ss += __shfl_xor(ss, 16, 32);
    mx = fmaxf(mx, __shfl_xor(mx, 16, 32));
    mn = fminf(mn, __shfl_xor(mn, 16, 32));
    if (half == 0) {
      atomicAdd(&statsf[col], s);
      atomicAdd(&statsf[256 + col], ss);
      atomicMax(&fmaxp[batch * 256 + col], __float_as_uint(mx));  // relu>=0: uint order == float order
      atomicMin(&fminp[batch * 256 + col], __float_as_uint(mn));
    }
  }
}

// ---------------------------------------------------------------- head: BNf pooled + 256->128->64->16 + log_softmax
__global__ __launch_bounds__(256) void head_kernel(const float* statsf, const unsigned* fmaxp,
                                                   const unsigned* fminp, const float* gf,
                                                   const float* bef, const float* wo1,
                                                   const float* bo1, const float* go1,
                                                   const float* beo1, const float* wo2,
                                                   const float* bo2, const float* go2,
                                                   const float* beo2, const float* wo3,
                                                   const float* bo3, float* out) {
  __shared__ float hp[BATCH][256];
  __shared__ float h1[BATCH][128];
  __shared__ float h2[BATCH][64];
  __shared__ float lg[BATCH][OUTC];
  int tid = threadIdx.x;
  {
    int c = tid;
    const float cnt = (float)NPTS;
    float m = statsf[c] / cnt;
    float v = statsf[256 + c] / cnt - m * m;
    float sc = gf[c] * rsqrtf(v + 1e-5f);
    float sh = bef[c] - m * sc;
    for (int b = 0; b < BATCH; ++b) {
      float mx = __uint_as_float(fmaxp[b * 256 + c]);
      float mn = __uint_as_float(fminp[b * 256 + c]);
      hp[b][c] = sc > 0.f ? sc * mx + sh : sc * mn + sh;
    }
  }
  __syncthreads();
  for (int t = tid; t < BATCH * 128; t += 256) {
    int b = t >> 7, c = t & 127;
    float acc = bo1[c];
    for (int d = 0; d < 256; ++d) acc += hp[b][d] * wo1[d * 128 + c];
    h1[b][c] = acc > 0.f ? acc : 0.f;
  }
  __syncthreads();
  if (tid < 128) {
    int c = tid; float s = 0.f, ss = 0.f;
    for (int b = 0; b < BATCH; ++b) { float z = h1[b][c]; s += z; ss += z * z; }
    float m = s / (float)BATCH, v = ss / (float)BATCH - m * m;
    float sc = go1[c] * rsqrtf(v + 1e-5f), sh = beo1[c] - m * sc;
    for (int b = 0; b < BATCH; ++b) h1[b][c] = sc * h1[b][c] + sh;
  }
  __syncthreads();
  for (int t = tid; t < BATCH * 64; t += 256) {
    int b = t >> 6, c = t & 63;
    float acc = bo2[c];
    for (int d = 0; d < 128; ++d) acc += h1[b][d] * wo2[d * 64 + c];
    h2[b][c] = acc > 0.f ? acc : 0.f;
  }
  __syncthreads();
  if (tid < 64) {
    int c = tid; float s = 0.f, ss = 0.f;
    for (int b = 0; b < BATCH; ++b) { float z = h2[b][c]; s += z; ss += z * z; }
    float m = s / (float)BATCH, v = ss / (float)BATCH - m * m;
    float sc = go2[c] * rsqrtf(v + 1e-5f), sh = beo2[c] - m * sc;
    for (int b = 0; b < BATCH; ++b) h2[b][c] = sc * h2[b][c] + sh;
  }
  __syncthreads();
  for (int t = tid; t < BATCH * OUTC; t += 256) {
    int b = t >> 4, o = t & 15;
    float acc = bo3[o];
    for (int d = 0; d < 64; ++d) acc += h2[b][d] * wo3[d * OUTC + o];
    lg[b][o] = acc;
  }
  __syncthreads();
  if (tid < BATCH) {
    int b = tid;
    float mx = -1e30f;
    for (int o = 0; o < OUTC; ++o) mx = fmaxf(mx, lg[b][o]);
    float s = 0.f;
    for (int o = 0; o < OUTC; ++o) s += __expf(lg[b][o] - mx);
    float lse = mx + __logf(s);
    for (int o = 0; o < OUTC; ++o) out[b * OUTC + o] = lg[b][o] - lse;
  }
}

// ---------------------------------------------------------------- launch
extern "C" void kernel_launch(void* const* d_in, const int* in_sizes, int n_in,
                              void* d_out, int out_size, void* d_ws, size_t ws_size,
                              hipStream_t stream) {
  const float* pos = (const float*)d_in[0];
  const float* x   = (const float*)d_in[1];
  // d_in[2] = batch (implicit: contiguous 2048-point groups)
  const float* w1 = (const float*)d_in[3];  const float* b1 = (const float*)d_in[4];
  const float* g1 = (const float*)d_in[5];  const float* be1 = (const float*)d_in[6];
  const float* w2 = (const float*)d_in[7];  const float* b2 = (const float*)d_in[8];
  const float* g2 = (const float*)d_in[9];  const float* be2 = (const float*)d_in[10];
  const float* wf = (const float*)d_in[11]; const float* bf = (const float*)d_in[12];
  const float* gf = (const float*)d_in[13]; const float* bef = (const float*)d_in[14];
  const float* wo1 = (const float*)d_in[15]; const float* bo1 = (const float*)d_in[16];
  const float* go1 = (const float*)d_in[17]; const float* beo1 = (const float*)d_in[18];
  const float* wo2 = (const float*)d_in[19]; const float* bo2 = (const float*)d_in[20];
  const float* go2 = (const float*)d_in[21]; const float* beo2 = (const float*)d_in[22];
  const float* wo3 = (const float*)d_in[23]; const float* bo3 = (const float*)d_in[24];
  float* out = (float*)d_out;

  float* wsf = (float*)d_ws;
  size_t off = 0;
  float* f0    = wsf + off; off += (size_t)NPTS * 4;
  float* sq    = wsf + off; off += NPTS;
  int*   idx1  = (int*)(wsf + off); off += (size_t)NPTS * KNBR;
  float* z1max = wsf + off; off += (size_t)NPTS * 64;
  float* z1min = wsf + off; off += (size_t)NPTS * 64;
  float* stats1 = wsf + off; off += 128;
  float* x1    = wsf + off; off += (size_t)NPTS * 64;
  float* x1sq  = wsf + off; off += NPTS;
  int*   idx2  = (int*)(wsf + off); off += (size_t)NPTS * KNBR;
  float* z2max = wsf + off; off += (size_t)NPTS * 128;
  float* z2min = wsf + off; off += (size_t)NPTS * 128;
  float* stats2 = wsf + off; off += 256;
  float* x2    = wsf + off; off += (size_t)NPTS * 128;
  float* statsf = wsf + off; off += 512;
  unsigned* fmaxp = (unsigned*)(wsf + off); off += BATCH * 256;
  unsigned* fminp = (unsigned*)(wsf + off); off += BATCH * 256;

  init_kernel<<<8, 256, 0, stream>>>(stats1, stats2, statsf, fmaxp, fminp);
  f0_kernel<<<NPTS / 256, 256, 0, stream>>>(pos, x, f0, sq);
  knn1_kernel<<<NPTS / 8, 256, 0, stream>>>(f0, sq, idx1);
  mlp1_kernel<<<NPTS / 8, 256, 0, stream>>>(f0, idx1, w1, b1, z1max, z1min, stats1);
  bn1_kernel<<<NPTS / 8, 256, 0, stream>>>(z1max, z1min, stats1, g1, be1, x1, x1sq);
  knn2_kernel<<<NPTS / 8, 256, 0, stream>>>(x1, x1sq, idx2);
  mlp2_kernel<<<NPTS, 256, 0, stream>>>(x1, idx2, w2, b2, z2max, z2min, stats2);
  bn2_kernel<<<(NPTS * 128) / 256, 256, 0, stream>>>(z2max, z2min, stats2, g2, be2, x2);
  mlpf_kernel<<<NPTS / 32, 256, 0, stream>>>(x1, x2, wf, bf, statsf, fmaxp, fminp);
  head_kernel<<<1, 256, 0, stream>>>(statsf, fmaxp, fminp, gf, bef, wo1, bo1, go1, beo1,
                                     wo2, bo2, go2, beo2, wo3, bo3, out);
}